// Model_67379446940240
// MI455X (gfx1250) — compile-verified
//
#include <hip/hip_runtime.h>

typedef float v2f __attribute__((ext_vector_type(2)));
typedef float v8f __attribute__((ext_vector_type(8)));

#define LOG2E_F 1.44269504088896f

static __device__ __forceinline__ float fast_exp2(float x) {
#if __has_builtin(__builtin_amdgcn_exp2f)
  return __builtin_amdgcn_exp2f(x);
#else
  return exp2f(x);
#endif
}
static __device__ __forceinline__ float fast_rcp(float x) {
#if __has_builtin(__builtin_amdgcn_rcpf)
  return __builtin_amdgcn_rcpf(x);
#else
  return 1.0f / x;
#endif
}
static __device__ __forceinline__ float fast_tanh(float x) {
#if __has_builtin(__builtin_amdgcn_tanhf)
  return __builtin_amdgcn_tanhf(x);       // v_tanh_f32 (confirmed in round-1 asm)
#else
  return __builtin_fmaf(2.0f, fast_rcp(1.0f + fast_exp2(x * (-2.0f * LOG2E_F))), -1.0f);
#endif
}
static __device__ __forceinline__ float fast_sigmoid(float x) {
  // sigmoid(x) == 0.5*tanh(0.5*x) + 0.5 exactly; 1 TRANS + 2 VALU (vs 2 TRANS + 2 VALU)
  return __builtin_fmaf(0.5f, fast_tanh(0.5f * x), 0.5f);
}

// Gate-row permutation: physical row m -> original gate row of W_ih/W_hh/b.
//   m in [0,16):  u = m>>2 (hidden 0..3), r = m&3 (0:i,1:f,2:g,3:o) -> orig = r*5+u
//   m in [16,20): u = 4,    r = m-16                               -> orig = r*5+4
// 16x16 f32 D layout (elem(m,n): lane = n + 16*(m>=8), vgpr = m%8):
//   lane n    holds rows 0..7  = ifgo(h0), ifgo(h1)  (tile1), rows 16..19 = ifgo(h4) (tile2)
//   lane n+16 holds rows 8..15 = ifgo(h2), ifgo(h3)
// B layout (4x16): vgpr0 = {k=0 | k=2}, vgpr1 = {k=1 | k=3} across lane halves, so
// B1 = {hA, hB}, B2 = {hC, 0} with per-lane hidden ownership (0,1,4 | 2,3) — zero shuffles.

__global__ __launch_bounds__(256) void lstm_wmma_kernel(
    const int* __restrict__ x,       // [65536, 256] int32 in {0,1}
    const float* __restrict__ emb,   // [2, 5]
    const float* __restrict__ W_ih,  // [20, 5]
    const float* __restrict__ W_hh,  // [20, 5]
    const float* __restrict__ b_ih,  // [20]
    const float* __restrict__ b_hh,  // [20]
    const float* __restrict__ W_lin, // [2, 5]
    const float* __restrict__ b_lin, // [2]
    float* __restrict__ out)         // [65536, 2]
{
  const int lane = threadIdx.x & 31;
  const int hi   = lane >> 4;        // 0: lanes 0-15, 1: lanes 16-31
  const int mm   = lane & 15;
  const int wave = blockIdx.x * (blockDim.x >> 5) + (threadIdx.x >> 5);
  const int nglob = wave * 16 + mm;  // batch owned by this lane (pair)

  // ---- Precompute per-lane gx variants (input contribution for x=0 / x=1) ----
  float gx0_t1[8], gx1_t1[8], gx0_t2[4], gx1_t2[4];
#pragma unroll
  for (int v = 0; v < 8; ++v) {
    int m    = v + 8 * hi;                 // physical row 0..15
    int orig = (m & 3) * 5 + (m >> 2);
    float bs = b_ih[orig] + b_hh[orig];
    float s0 = bs, s1 = bs;
#pragma unroll
    for (int e = 0; e < 5; ++e) {
      float w = W_ih[orig * 5 + e];
      s0 = __builtin_fmaf(emb[e],     w, s0);
      s1 = __builtin_fmaf(emb[5 + e], w, s1);
    }
    gx0_t1[v] = s0;
    gx1_t1[v] = s1;
  }
#pragma unroll
  for (int v = 0; v < 4; ++v) {
    int orig = v * 5 + 4;                  // physical row 16+v
    float bs = b_ih[orig] + b_hh[orig];
    float s0 = bs, s1 = bs;
#pragma unroll
    for (int e = 0; e < 5; ++e) {
      float w = W_ih[orig * 5 + e];
      s0 = __builtin_fmaf(emb[e],     w, s0);
      s1 = __builtin_fmaf(emb[5 + e], w, s1);
    }
    gx0_t2[v] = hi ? 0.0f : s0;            // lanes>=16: tile2 rows are dead -> 0
    gx1_t2[v] = hi ? 0.0f : s1;
  }

  // ---- Load A matrices (W_hh, permuted rows) in 16x4 f32 A layout ----
  const int kb    = hi ? 2 : 0;            // K pair held by this lane half
  const int origA = (mm & 3) * 5 + (mm >> 2);   // row P(mm), mm in 0..15
  v2f a1a, a1b, a2a, a2b;
  a1a.x = W_hh[origA * 5 + kb];
  a1a.y = W_hh[origA * 5 + kb + 1];
  if (mm < 4) {                            // tile2 valid rows 0..3 = phys 16..19
    int ob = mm * 5 + 4;
    a1b.x = W_hh[ob * 5 + kb];
    a1b.y = W_hh[ob * 5 + kb + 1];
  } else {
    a1b.x = 0.0f; a1b.y = 0.0f;
  }
  a2a.x = hi ? 0.0f : W_hh[origA * 5 + 4]; // K-group2 col0 = hidden 4, cols1-3 = 0
  a2a.y = 0.0f;
  a2b.x = (!hi && mm < 4) ? W_hh[(mm * 5 + 4) * 5 + 4] : 0.0f;
  a2b.y = 0.0f;

  // ---- Bit-pack this batch's 256 tokens into 8 words ----
  unsigned words[8];
  const int4* xv = reinterpret_cast<const int4*>(x + (size_t)nglob * 256);
#pragma unroll
  for (int w8 = 0; w8 < 8; ++w8) {
    unsigned bits = 0u;
#pragma unroll
    for (int q = 0; q < 8; ++q) {
      int4 v = xv[w8 * 8 + q];
      bits |= (unsigned)(v.x & 1) << (q * 4 + 0);
      bits |= (unsigned)(v.y & 1) << (q * 4 + 1);
      bits |= (unsigned)(v.z & 1) << (q * 4 + 2);
      bits |= (unsigned)(v.w & 1) << (q * 4 + 3);
    }
    words[w8] = bits;
  }

  // ---- Recurrence: lane owns hidden (0,1,4) or (2,3) of its batch ----
  float hA = 0.0f, hB = 0.0f, hC = 0.0f;
  float cA = 0.0f, cB = 0.0f, cC = 0.0f;

  // Persistent tile-2 accumulator: elements 4..7 are written once (0) and then
  // only ever pass through WMMAs whose corresponding A rows are all-zero, so no
  // per-iteration re-zeroing is required.
  v8f d2 = {};

  for (int wI = 0; wI < 8; ++wI) {
    unsigned wbits = words[wI];
#pragma unroll 8
    for (int j = 0; j < 32; ++j) {
      bool xb = (wbits >> j) & 1u;

      v8f c1;
#pragma unroll
      for (int v = 0; v < 8; ++v) c1[v] = xb ? gx1_t1[v] : gx0_t1[v];
#pragma unroll
      for (int v = 0; v < 4; ++v) d2[v] = xb ? gx1_t2[v] : gx0_t2[v];

      v2f b1; b1.x = hA; b1.y = hB;        // k rows {0|2},{1|3}
      v2f b2; b2.x = hC; b2.y = 0.0f;      // k rows {4|zero},{zero}

      // gates^T = W_hh_perm @ h^T + gx_t  (K=5 split as 4+1, N=16 batches)
      v8f d1 = __builtin_amdgcn_wmma_f32_16x16x4_f32(false, a1a, false, b1,
                                                     (short)0, c1, false, false);
      d1     = __builtin_amdgcn_wmma_f32_16x16x4_f32(false, a2a, false, b2,
                                                     (short)0, d1, false, false);
      d2     = __builtin_amdgcn_wmma_f32_16x16x4_f32(false, a1b, false, b1,
                                                     (short)0, d2, false, false);
      d2     = __builtin_amdgcn_wmma_f32_16x16x4_f32(false, a2b, false, b2,
                                                     (short)0, d2, false, false);

      // Elementwise LSTM update, no cross-lane traffic.
      float iA = fast_sigmoid(d1[0]), fA = fast_sigmoid(d1[1]);
      float gA = fast_tanh(d1[2]),    oA = fast_sigmoid(d1[3]);
      cA = __builtin_fmaf(fA, cA, iA * gA);
      hA = oA * fast_tanh(cA);

      float iB = fast_sigmoid(d1[4]), fB = fast_sigmoid(d1[5]);
      float gB = fast_tanh(d1[6]),    oB = fast_sigmoid(d1[7]);
      cB = __builtin_fmaf(fB, cB, iB * gB);
      hB = oB * fast_tanh(cB);

      float iC = fast_sigmoid(d2[0]), fC = fast_sigmoid(d2[1]);
      float gC = fast_tanh(d2[2]),    oC = fast_sigmoid(d2[3]);
      cC = __builtin_fmaf(fC, cC, iC * gC);
      float hCn = oC * fast_tanh(cC);
      hC = hi ? 0.0f : hCn;                // keep B2 row k=2 slot exactly zero
    }
  }

  // ---- Head: out[n, o] = h . W_lin[o, :] + b_lin[o]; reduce across lane pair ----
  const int col0 = hi ? 2 : 0;
  const int col1 = hi ? 3 : 1;
  float p0 = hA * W_lin[col0] + hB * W_lin[col1] + hC * W_lin[4];      // hC==0 on hi
  float p1 = hA * W_lin[5 + col0] + hB * W_lin[5 + col1] + hC * W_lin[9];
  p0 += __shfl_xor(p0, 16, 32);
  p1 += __shfl_xor(p1, 16, 32);
  if (!hi) {
    float2 r;
    r.x = p0 + b_lin[0];
    r.y = p1 + b_lin[1];
    reinterpret_cast<float2*>(out)[nglob] = r;
  }
}

extern "C" void kernel_launch(void* const* d_in, const int* in_sizes, int n_in,
                              void* d_out, int out_size, void* d_ws, size_t ws_size,
                              hipStream_t stream) {
  (void)in_sizes; (void)n_in; (void)out_size; (void)d_ws; (void)ws_size;
  const int*   x     = (const int*)  d_in[0];
  const float* emb   = (const float*)d_in[1];
  const float* W_ih  = (const float*)d_in[2];
  const float* W_hh  = (const float*)d_in[3];
  const float* b_ih  = (const float*)d_in[4];
  const float* b_hh  = (const float*)d_in[5];
  const float* W_lin = (const float*)d_in[6];
  const float* b_lin = (const float*)d_in[7];
  float* out = (float*)d_out;

  // 65536 batches / 16 per wave / 8 waves per block = 512 blocks
  dim3 grid(512), block(256);
  lstm_wmma_kernel<<<grid, block, 0, stream>>>(x, emb, W_ih, W_hh, b_ih, b_hh,
                                               W_lin, b_lin, out);
}